// GateCriterionModel_73813307949239
// MI455X (gfx1250) — compile-verified
//
#include <hip/hip_runtime.h>

// ---------------------------------------------------------------------------
// Problem constants (match reference)
// ---------------------------------------------------------------------------
constexpr int Bc  = 128;   // batch
constexpr int Lc  = 256;   // sequence length
constexpr int Hc  = 128;   // hidden
constexpr int H2c = 256;   // 2*H
constexpr int VSc = 256;   // vocab
constexpr float THRESH2 = 0.4f * 0.4f;

typedef __attribute__((ext_vector_type(2))) float v2f;
typedef __attribute__((ext_vector_type(8))) float v8f;

// D = A(16x4) * B(4x16) + C, all f32, exact f32 math on the matrix pipe.
__device__ __forceinline__ v8f wmma_f32(v2f a, v2f b, v8f c) {
  return __builtin_amdgcn_wmma_f32_16x16x4_f32(
      /*neg_a=*/false, a, /*neg_b=*/false, b,
      /*c_mod=*/(short)0, c, /*reuse_a=*/false, /*reuse_b=*/false);
}

// ---------------------------------------------------------------------------
// Kernel 1: fused  embed-gather -> FFN -> residual -> LayerNorm -> K = h'Wk^T
// One wave handles a 16-token tile end-to-end. Weights streamed from global
// (L2-resident: reused by all 2048 tiles). Tile data staged in LDS.
// block = 128 threads (4 waves), dynamic LDS = 4 * (16*128+16*256+16*128) * 4B
// ---------------------------------------------------------------------------
__global__ void fused_token_kernel(const int*   __restrict__ seq,
                                   const float* __restrict__ embed,
                                   const float* __restrict__ W1,
                                   const float* __restrict__ b1,
                                   const float* __restrict__ W2,
                                   const float* __restrict__ b2,
                                   const float* __restrict__ gamma,
                                   const float* __restrict__ beta,
                                   const float* __restrict__ Wk,
                                   float*       __restrict__ Kout) {
  extern __shared__ float smem[];
  const int lane = threadIdx.x & 31;
  const int wave = threadIdx.x >> 5;

  float* hbuf = smem + wave * (16 * Hc + 16 * H2c + 16 * Hc); // 16 x 128 (h, later h')
  float* tbuf = hbuf + 16 * Hc;                               // 16 x 256 (relu act)
  float* xbuf = tbuf + 16 * H2c;                              // 16 x 128 (residual x)

  const int tile = blockIdx.x * 4 + wave;
  const int tok0 = tile * 16;

  const int half = lane >> 4;   // 0: K{0,1}, 1: K{2,3} slice of the fragment
  const int lr   = lane & 15;   // row (A) / col (B,D) within the 16-wide tile

  // ---- gather embeddings: hbuf[r][c] = embed[seq[tok0+r]][c] ----
  for (int r = 0; r < 16; ++r) {
    const int s = seq[tok0 + r];
    const float* erow = embed + (size_t)s * Hc;
    for (int c = lane; c < Hc; c += 32) hbuf[r * Hc + c] = erow[c];
  }

  // ---- GEMM1: t = relu(h @ W1^T + b1)   (16 x 256), K-dim = 128 ----
  for (int nt = 0; nt < H2c / 16; ++nt) {
    const int n0 = nt * 16;
    v8f acc = {};
    for (int k0 = 0; k0 < Hc; k0 += 4) {
      v2f a = *(const v2f*)(hbuf + lr * Hc + k0 + 2 * half);
      v2f b = *(const v2f*)(W1 + (size_t)(n0 + lr) * Hc + k0 + 2 * half);
      acc = wmma_f32(a, b, acc);
    }
    const float bias = b1[n0 + lr];
#pragma unroll
    for (int r = 0; r < 8; ++r) {
      const int row = r + 8 * half;
      const float v = acc[r] + bias;
      tbuf[row * H2c + n0 + lr] = v > 0.f ? v : 0.f;
    }
  }

  // ---- GEMM2 + residual: x = h + t @ W2^T + b2   (16 x 128), K-dim = 256 ----
  for (int nt = 0; nt < Hc / 16; ++nt) {
    const int n0 = nt * 16;
    v8f acc = {};
    for (int k0 = 0; k0 < H2c; k0 += 4) {
      v2f a = *(const v2f*)(tbuf + lr * H2c + k0 + 2 * half);
      v2f b = *(const v2f*)(W2 + (size_t)(n0 + lr) * H2c + k0 + 2 * half);
      acc = wmma_f32(a, b, acc);
    }
    const float bias = b2[n0 + lr];
#pragma unroll
    for (int r = 0; r < 8; ++r) {
      const int row = r + 8 * half;
      xbuf[row * Hc + n0 + lr] = acc[r] + bias + hbuf[row * Hc + n0 + lr];
    }
  }

  // ---- LayerNorm per row (biased var, eps=1e-5) -> hbuf = h' ----
  for (int r = 0; r < 16; ++r) {
    float s = 0.f, sq = 0.f;
    for (int c = lane; c < Hc; c += 32) {
      const float v = xbuf[r * Hc + c];
      s += v; sq += v * v;
    }
    for (int m = 16; m > 0; m >>= 1) {
      s  += __shfl_xor(s,  m, 32);
      sq += __shfl_xor(sq, m, 32);
    }
    const float mu  = s * (1.f / Hc);
    const float var = sq * (1.f / Hc) - mu * mu;
    const float inv = rsqrtf(var + 1e-5f);
    for (int c = lane; c < Hc; c += 32)
      hbuf[r * Hc + c] = (xbuf[r * Hc + c] - mu) * inv * gamma[c] + beta[c];
  }

  // ---- GEMM3: K = h' @ Wk^T  (16 x 128) -> global ----
  for (int nt = 0; nt < Hc / 16; ++nt) {
    const int n0 = nt * 16;
    v8f acc = {};
    for (int k0 = 0; k0 < Hc; k0 += 4) {
      v2f a = *(const v2f*)(hbuf + lr * Hc + k0 + 2 * half);
      v2f b = *(const v2f*)(Wk + (size_t)(n0 + lr) * Hc + k0 + 2 * half);
      acc = wmma_f32(a, b, acc);
    }
#pragma unroll
    for (int r = 0; r < 8; ++r)
      Kout[(size_t)(tok0 + r + 8 * half) * Hc + n0 + lr] = acc[r];
  }
}

// ---------------------------------------------------------------------------
// Kernel 2: sequential gated delta-rule scan. One workgroup per batch element;
// the 128x128 state M lives entirely in LDS (stride 129 to spread banks).
// 255 sequential steps: matvec (2 thr/row), gate test, rank-1 update.
// block = 256 threads, dynamic LDS = (128*129 + 256 + 3*128) * 4 bytes.
// ---------------------------------------------------------------------------
__global__ void delta_scan_kernel(const float* __restrict__ Kin,
                                  float*       __restrict__ readOut) {
  extern __shared__ float smem[];
  float* M    = smem;                 // 128 x 129 (padded)
  float* red  = M + 128 * 129;        // 256
  float* kvec = red + 256;            // 128
  float* knrm = kvec + 128;           // 128 (normalized k)
  float* errv = knrm + 128;           // 128

  const int tid = threadIdx.x;
  const int b   = blockIdx.x;
  const float* Kb = Kin + (size_t)b * Lc * Hc;

  for (int i = tid; i < 128 * 129; i += 256) M[i] = 0.f;
  __syncthreads();

  const int row  = tid >> 1;
  const int hq   = tid & 1;           // which half of the dot product

  for (int l = 0; l < Lc - 1; ++l) {
    const float* kp = Kb + l * Hc;
    const float kv = (tid < Hc) ? kp[tid] : 0.f;

    // ||k||^2
    red[tid] = kv * kv;
    __syncthreads();
    for (int s = 128; s > 0; s >>= 1) {
      if (tid < s) red[tid] += red[tid + s];
      __syncthreads();
    }
    const float knorm2 = red[0];
    __syncthreads();
    const float inv = 1.f / fmaxf(sqrtf(knorm2), 1e-12f);
    if (tid < Hc) { kvec[tid] = kv; knrm[tid] = kv * inv; }
    __syncthreads();

    // vp = M @ kn  (2 threads per row, 64 MACs each)
    {
      const float* Mr = M + row * 129 + hq * 64;
      const float* kh = knrm + hq * 64;
      float part = 0.f;
#pragma unroll 8
      for (int j = 0; j < 64; ++j) part += Mr[j] * kh[j];
      red[tid] = part;
    }
    __syncthreads();
    if (tid < Hc) {
      const float vp = red[2 * tid] + red[2 * tid + 1];
      errv[tid] = kvec[tid] - vp;
    }
    __syncthreads();

    // ||err||^2 and gate
    red[tid] = (tid < Hc) ? errv[tid] * errv[tid] : 0.f;
    __syncthreads();
    for (int s = 128; s > 0; s >>= 1) {
      if (tid < s) red[tid] += red[tid + s];
      __syncthreads();
    }
    const bool gate = (red[0] >= THRESH2 * knorm2);   // ||e|| >= 0.4*||k||
    __syncthreads();

    // M += err (outer) kn   (uniform branch per workgroup)
    if (gate) {
      const int base = tid * 64;
#pragma unroll 4
      for (int j = 0; j < 64; ++j) {
        const int idx = base + j;
        const int r2 = idx >> 7;
        const int c2 = idx & 127;
        M[r2 * 129 + c2] += errv[r2] * knrm[c2];
      }
    }
    __syncthreads();
  }

  // read = M @ q, q = K[:, L-1, :]
  const float* qp = Kb + (size_t)(Lc - 1) * Hc;
  if (tid < Hc) knrm[tid] = qp[tid];
  __syncthreads();
  {
    const float* Mr = M + row * 129 + hq * 64;
    const float* kh = knrm + hq * 64;
    float part = 0.f;
#pragma unroll 8
    for (int j = 0; j < 64; ++j) part += Mr[j] * kh[j];
    red[tid] = part;
  }
  __syncthreads();
  if (tid < Hc) readOut[(size_t)b * Hc + tid] = red[2 * tid] + red[2 * tid + 1];
}

// ---------------------------------------------------------------------------
// Kernel 3: generic WMMA tile GEMM  C = A(MxK) @ Bw(NxK)^T + bias
// One wave per 16x16 output tile; 4 waves per block.
// ---------------------------------------------------------------------------
__global__ void gemm_bias_kernel(const float* __restrict__ A,
                                 const float* __restrict__ Bw,
                                 const float* __restrict__ bias,
                                 float*       __restrict__ C,
                                 int Ndim, int Kdim) {
  const int lane = threadIdx.x & 31;
  const int wave = threadIdx.x >> 5;
  const int tile = blockIdx.x * (blockDim.x >> 5) + wave;
  const int ntiles = Ndim >> 4;
  const int m0 = (tile / ntiles) * 16;
  const int n0 = (tile % ntiles) * 16;
  const int half = lane >> 4;
  const int lr   = lane & 15;

  v8f acc = {};
  for (int k0 = 0; k0 < Kdim; k0 += 4) {
    v2f a = *(const v2f*)(A  + (size_t)(m0 + lr) * Kdim + k0 + 2 * half);
    v2f b = *(const v2f*)(Bw + (size_t)(n0 + lr) * Kdim + k0 + 2 * half);
    acc = wmma_f32(a, b, acc);
  }
  const float bv = bias[n0 + lr];
#pragma unroll
  for (int r = 0; r < 8; ++r)
    C[(size_t)(m0 + r + 8 * half) * Ndim + n0 + lr] = acc[r] + bv;
}

// ---------------------------------------------------------------------------
extern "C" void kernel_launch(void* const* d_in, const int* in_sizes, int n_in,
                              void* d_out, int out_size, void* d_ws, size_t ws_size,
                              hipStream_t stream) {
  (void)in_sizes; (void)n_in; (void)out_size; (void)ws_size;

  const int*   seq   = (const int*)  d_in[0];
  const float* embed = (const float*)d_in[1];
  const float* W1    = (const float*)d_in[2];
  const float* b1    = (const float*)d_in[3];
  const float* W2    = (const float*)d_in[4];
  const float* b2    = (const float*)d_in[5];
  const float* gamma = (const float*)d_in[6];
  const float* beta  = (const float*)d_in[7];
  const float* Wk    = (const float*)d_in[8];
  const float* Wr    = (const float*)d_in[9];
  const float* br    = (const float*)d_in[10];
  const float* Wo    = (const float*)d_in[11];
  const float* bo    = (const float*)d_in[12];
  float* out = (float*)d_out;

  // workspace layout
  float* Kbuf    = (float*)d_ws;                            // B*L*H   = 16 MB
  float* readBuf = Kbuf    + (size_t)Bc * Lc * Hc;          // B*H
  float* r2Buf   = readBuf + (size_t)Bc * Hc;               // B*H

  // 1) fused token pipeline -> K
  {
    const int tiles = (Bc * Lc) / 16;                       // 2048
    const int blocks = tiles / 4;                           // 512
    const size_t lds = 4u * (16 * Hc + 16 * H2c + 16 * Hc) * sizeof(float); // 128 KB
    fused_token_kernel<<<blocks, 128, lds, stream>>>(
        seq, embed, W1, b1, W2, b2, gamma, beta, Wk, Kbuf);
  }

  // 2) sequential gated delta-rule scan (M resident in LDS per batch)
  {
    const size_t lds = (size_t)(128 * 129 + 256 + 3 * 128) * sizeof(float); // ~67 KB
    delta_scan_kernel<<<Bc, 256, lds, stream>>>(Kbuf, readBuf);
  }

  // 3) r2 = read @ Wr^T + br  (128x128, K=128) -> 64 tiles -> 16 blocks
  gemm_bias_kernel<<<(Bc / 16) * (Hc / 16) / 4, 128, 0, stream>>>(
      readBuf, Wr, br, r2Buf, Hc, Hc);

  // 4) out = r2 @ Wo^T + bo   (128x256, K=128) -> 128 tiles -> 32 blocks
  gemm_bias_kernel<<<(Bc / 16) * (VSc / 16) / 4, 128, 0, stream>>>(
      r2Buf, Wo, bo, out, VSc, Hc);
}